// BELayer_21844203667566
// MI455X (gfx1250) — compile-verified
//
#include <hip/hip_runtime.h>
#include <math.h>

#define N_S   256
#define D1    512
#define D2    2048
#define EPS   1e-5f
#define BLOCK 512
#define WAVES 16            // BLOCK / 32 (wave32)
#define STRIP 64            // e-columns per wave per strip (4 WMMA tiles)
#define STRIPS_PER_WAVE (D2 / (STRIP * WAVES))   // = 2

typedef __attribute__((ext_vector_type(2))) float v2f;
typedef __attribute__((ext_vector_type(8))) float v8f;

__device__ __forceinline__ float wave_reduce_add(float v) {
    #pragma unroll
    for (int off = 16; off > 0; off >>= 1)
        v += __shfl_xor(v, off, 32);
    return v;
}

__global__ __launch_bounds__(BLOCK) void belayer_kernel(
    const float* __restrict__ x,    const float* __restrict__ U,
    const float* __restrict__ bias, const float* __restrict__ g1,
    const float* __restrict__ b1,   const float* __restrict__ g2,
    const float* __restrict__ b2,   float* __restrict__ out)
{
    const int n    = blockIdx.x;
    const int tid  = threadIdx.x;
    const int lane = tid & 31;
    const int wave = tid >> 5;
    const int half = lane >> 4;     // 0: lanes 0-15 (K=0,1), 1: lanes 16-31 (K=2,3)
    const int ln16 = lane & 15;

    __shared__ float s_h[D1];
    __shared__ float s_act[D2];
    __shared__ float s_red[2 * WAVES];
    __shared__ float s_stats[2];

    // ---------- Phase 1: h = LayerNorm(x[n]) -> LDS ----------
    const float xv = x[(size_t)n * D1 + tid];          // BLOCK == D1
    {
        float s  = wave_reduce_add(xv);
        float sq = wave_reduce_add(xv * xv);
        if (lane == 0) { s_red[wave] = s; s_red[WAVES + wave] = sq; }
    }
    __syncthreads();
    if (wave == 0) {
        float a = (lane < WAVES) ? s_red[lane]         : 0.0f;
        float b = (lane < WAVES) ? s_red[WAVES + lane] : 0.0f;
        a = wave_reduce_add(a);
        b = wave_reduce_add(b);
        if (lane == 0) {
            float mu  = a * (1.0f / D1);
            float var = b * (1.0f / D1) - mu * mu;
            s_stats[0] = mu;
            s_stats[1] = rsqrtf(var + EPS);
        }
    }
    __syncthreads();
    s_h[tid] = (xv - s_stats[0]) * s_stats[1] * g1[tid] + b1[tid];
    __syncthreads();

    // ---------- Phase 2: act = h . U[n] + bias via V_WMMA_F32_16X16X4_F32 ----------
    // B(4x16) f32 layout: vgpr0 = {lanes0-15:K0, lanes16-31:K2}, vgpr1 = {K1, K3}
    // A(16x4) f32 layout: vgpr0 = {rows, K0|K2}, vgpr1 = {rows, K1|K3}; all 16 rows
    // carry the same h chunk, so D row 0 (acc[0] of lanes 0-15) is the dot-product.
    const float* __restrict__ Un = U + (size_t)n * D1 * D2;

    #pragma unroll
    for (int s = 0; s < STRIPS_PER_WAVE; ++s) {
        const int e0 = (wave + s * WAVES) * STRIP;
        const float* __restrict__ Ub = Un + e0 + ln16;   // per-lane column base

        v8f acc0 = {}, acc1 = {}, acc2 = {}, acc3 = {};

        #pragma unroll 4
        for (int d0 = 0; d0 < D1; d0 += 4) {
            const int kr = d0 + 2 * half;                 // K row for this half-wave
            v2f a;
            a.x = s_h[kr];                                // K = 0 | 2
            a.y = s_h[kr + 1];                            // K = 1 | 3

            if (d0 < D1 - 20)                             // stream-ahead into GL2/L0
                __builtin_prefetch(Ub + (size_t)(kr + 16) * D2, 0, 0);

            const float* __restrict__ p0 = Ub + (size_t)kr * D2;
            v2f b0, b1v, b2v, b3;
            b0.x  = p0[0];        b0.y  = p0[D2];
            b1v.x = p0[16];       b1v.y = p0[D2 + 16];
            b2v.x = p0[32];       b2v.y = p0[D2 + 32];
            b3.x  = p0[48];       b3.y  = p0[D2 + 48];

            acc0 = __builtin_amdgcn_wmma_f32_16x16x4_f32(false, a, false, b0,  (short)0, acc0, false, false);
            acc1 = __builtin_amdgcn_wmma_f32_16x16x4_f32(false, a, false, b1v, (short)0, acc1, false, false);
            acc2 = __builtin_amdgcn_wmma_f32_16x16x4_f32(false, a, false, b2v, (short)0, acc2, false, false);
            acc3 = __builtin_amdgcn_wmma_f32_16x16x4_f32(false, a, false, b3,  (short)0, acc3, false, false);
        }

        if (half == 0) {                                  // lanes 0-15 hold D row M=0
            const size_t bb = (size_t)n * D2 + e0 + ln16;
            s_act[e0 + ln16     ] = acc0[0] + bias[bb     ];
            s_act[e0 + ln16 + 16] = acc1[0] + bias[bb + 16];
            s_act[e0 + ln16 + 32] = acc2[0] + bias[bb + 32];
            s_act[e0 + ln16 + 48] = acc3[0] + bias[bb + 48];
        }
    }
    __syncthreads();

    // ---------- Phase 3: LayerNorm over e (2048) + exact-erf GELU ----------
    float v[D2 / BLOCK];
    float ls = 0.0f, lq = 0.0f;
    #pragma unroll
    for (int i = 0; i < D2 / BLOCK; ++i) {
        v[i] = s_act[tid + i * BLOCK];
        ls += v[i];
        lq += v[i] * v[i];
    }
    {
        float s  = wave_reduce_add(ls);
        float sq = wave_reduce_add(lq);
        if (lane == 0) { s_red[wave] = s; s_red[WAVES + wave] = sq; }
    }
    __syncthreads();
    if (wave == 0) {
        float a = (lane < WAVES) ? s_red[lane]         : 0.0f;
        float b = (lane < WAVES) ? s_red[WAVES + lane] : 0.0f;
        a = wave_reduce_add(a);
        b = wave_reduce_add(b);
        if (lane == 0) {
            float mu  = a * (1.0f / D2);
            float var = b * (1.0f / D2) - mu * mu;
            s_stats[0] = mu;
            s_stats[1] = rsqrtf(var + EPS);
        }
    }
    __syncthreads();
    {
        const float mu = s_stats[0], rs = s_stats[1];
        #pragma unroll
        for (int i = 0; i < D2 / BLOCK; ++i) {
            const int e = tid + i * BLOCK;
            const float y = (v[i] - mu) * rs * g2[e] + b2[e];
            out[(size_t)n * D2 + e] = 0.5f * y * (1.0f + erff(y * 0.70710678118654752f));
        }
    }
}

extern "C" void kernel_launch(void* const* d_in, const int* in_sizes, int n_in,
                              void* d_out, int out_size, void* d_ws, size_t ws_size,
                              hipStream_t stream) {
    const float* x    = (const float*)d_in[0];
    const float* U    = (const float*)d_in[1];
    const float* bias = (const float*)d_in[2];
    const float* g1   = (const float*)d_in[3];
    const float* b1   = (const float*)d_in[4];
    const float* g2   = (const float*)d_in[5];
    const float* b2   = (const float*)d_in[6];
    float* out = (float*)d_out;
    hipLaunchKernelGGL(belayer_kernel, dim3(N_S), dim3(BLOCK), 0, stream,
                       x, U, bias, g1, b1, g2, b2, out);
}